// OurGMNCustomInter_8924942041964
// MI455X (gfx1250) — compile-verified
//
#include <hip/hip_runtime.h>
#include <hip/hip_bf16.h>

// Problem constants (fixed by the reference).
#define NQ  20000
#define NT  20000
#define NE  640000
#define DD  256

typedef __attribute__((ext_vector_type(2))) float v2f;
typedef __attribute__((ext_vector_type(8))) float v8f;

// ---------------------------------------------------------------------------
// order-preserving float <-> uint mapping (for atomicMax on floats)
__device__ __forceinline__ unsigned f2ord(float f) {
    unsigned b = __float_as_uint(f);
    return (b & 0x80000000u) ? ~b : (b | 0x80000000u);
}
__device__ __forceinline__ float ord2f(unsigned o) {
    unsigned b = (o & 0x80000000u) ? (o & 0x7FFFFFFFu) : ~o;
    return __uint_as_float(b);
}

// ---------------------------------------------------------------------------
// One K=256 pass of V_WMMA_F32_16X16X4_F32 accumulation.
//   Ap : this lane's A row (row stride already applied; broadcast row for Q)
//   Wc : W + kbase*DD + col   (column pointer into the weight block)
// Straight-line: 4 WMMAs per 16-wide K chunk, loads batched ahead of the MMAs.
__device__ __forceinline__ v8f mma_k256(const float* __restrict__ Ap,
                                        const float* __restrict__ Wc,
                                        int khalf, v8f acc)
{
#pragma unroll 4
    for (int k = 0; k < DD; k += 16) {
        float2 a0 = *(const float2*)(Ap + k + khalf);        // K = k+khalf, +1
        float2 a1 = *(const float2*)(Ap + k + 4  + khalf);
        float2 a2 = *(const float2*)(Ap + k + 8  + khalf);
        float2 a3 = *(const float2*)(Ap + k + 12 + khalf);
        const float* w0 = Wc + (long)(k + khalf) * DD;
        float b0x = w0[0],         b0y = w0[DD];
        float b1x = w0[4  * DD],   b1y = w0[5  * DD];
        float b2x = w0[8  * DD],   b2y = w0[9  * DD];
        float b3x = w0[12 * DD],   b3y = w0[13 * DD];
        __builtin_prefetch(w0 + 16 * DD, 0, 1);  // speculative: OOB is dropped by HW

        v2f a, b;
        a.x = a0.x; a.y = a0.y; b.x = b0x; b.y = b0y;
        acc = __builtin_amdgcn_wmma_f32_16x16x4_f32(false, a, false, b, (short)0, acc, false, false);
        a.x = a1.x; a.y = a1.y; b.x = b1x; b.y = b1y;
        acc = __builtin_amdgcn_wmma_f32_16x16x4_f32(false, a, false, b, (short)0, acc, false, false);
        a.x = a2.x; a.y = a2.y; b.x = b2x; b.y = b2y;
        acc = __builtin_amdgcn_wmma_f32_16x16x4_f32(false, a, false, b, (short)0, acc, false, false);
        a.x = a3.x; a.y = a3.y; b.x = b3x; b.y = b3y;
        acc = __builtin_amdgcn_wmma_f32_16x16x4_f32(false, a, false, b, (short)0, acc, false, false);
    }
    return acc;
}

// ---------------------------------------------------------------------------
// WMMA f32 GEMM:  C[M,256] = act( [A0 | A1? | A2row?][M, 256*(1+HAS1+HAS2)] @ W + bias )
// One wave per 16x16 output tile. Every A-part is exactly DD=256 wide, so the
// part selection is compile-time (no branches in the K loop). A2 is a single
// broadcast row (the mean-Q part of the t-merge).
template<bool HAS1, bool HAS2, bool ELU>
__global__ __launch_bounds__(256) void gemm_wmma_f32(
    const float* __restrict__ A0, const float* __restrict__ A1,
    const float* __restrict__ A2row,
    const float* __restrict__ W, const float* __restrict__ bias,
    float* __restrict__ C)
{
    const int wave = (blockIdx.x * blockDim.x + threadIdx.x) >> 5;
    const int lane = threadIdx.x & 31;
    const int tm = wave >> 4;              // DD/16 == 16 tiles along N
    const int tn = wave & 15;
    const int row   = tm * 16 + (lane & 15);   // A-frag row for this lane
    const int col   = tn * 16 + (lane & 15);   // B/C column for this lane
    const int khalf = (lane >> 4) << 1;        // 0 (lanes 0-15) / 2 (lanes 16-31)

    v8f acc = {};
    acc = mma_k256(A0 + (long)row * DD, W + col, khalf, acc);
    if (HAS1) acc = mma_k256(A1 + (long)row * DD, W + (long)DD * DD     + col, khalf, acc);
    if (HAS2) acc = mma_k256(A2row,               W + (long)2 * DD * DD + col, khalf, acc);

    const float bv = bias[col];
#pragma unroll
    for (int r = 0; r < 8; ++r) {
        const int orow = tm * 16 + r + ((lane >> 4) << 3); // M = r (lanes<16) / r+8
        float x = acc[r] + bv;
        if (ELU) x = (x > 0.f) ? x : expm1f(x);            // jax.nn.elu, alpha=1
        C[(long)orow * DD + col] = x;
    }
}

// ---------------------------------------------------------------------------
// One wave per edge: 256-wide dot product + fused segment-max (atomicMax on
// order-mapped bits). Lane i owns channels [8i, 8i+8) -> fully coalesced 1KB/row.
__global__ __launch_bounds__(256) void edge_logits(
    const float* __restrict__ Xq_att, const float* __restrict__ Xt_att,
    const int* __restrict__ u_idx, const int* __restrict__ v_idx,
    float* __restrict__ logits, unsigned* __restrict__ umax, unsigned* __restrict__ vmax)
{
    const int e    = (blockIdx.x * blockDim.x + threadIdx.x) >> 5;
    const int lane = threadIdx.x & 31;
    if (e >= NE) return;
    const int u = u_idx[e], v = v_idx[e];

    const float4* qa = (const float4*)(Xq_att + (long)u * DD) + lane * 2;
    const float4* ta = (const float4*)(Xt_att + (long)v * DD) + lane * 2;
    float4 q0 = qa[0], q1 = qa[1], t0 = ta[0], t1 = ta[1];
    float s = q0.x*t0.x + q0.y*t0.y + q0.z*t0.z + q0.w*t0.w
            + q1.x*t1.x + q1.y*t1.y + q1.z*t1.z + q1.w*t1.w;
#pragma unroll
    for (int m = 16; m >= 1; m >>= 1) s += __shfl_xor(s, m, 32);

    if (lane == 0) {
        logits[e] = s;
        unsigned o = f2ord(s);
        atomicMax(&umax[u], o);
        atomicMax(&vmax[v], o);
    }
}

// exp(l - segmax) for both segmentations + segment sums.
__global__ __launch_bounds__(256) void edge_expsum(
    const float* __restrict__ logits,
    const int* __restrict__ u_idx, const int* __restrict__ v_idx,
    const unsigned* __restrict__ umax, const unsigned* __restrict__ vmax,
    float* __restrict__ eu_arr, float* __restrict__ ev_arr,
    float* __restrict__ usum, float* __restrict__ vsum)
{
    const int e = blockIdx.x * blockDim.x + threadIdx.x;
    if (e >= NE) return;
    const float l = logits[e];
    const int u = u_idx[e], v = v_idx[e];
    const float eu = __expf(l - ord2f(umax[u]));
    const float ev = __expf(l - ord2f(vmax[v]));
    eu_arr[e] = eu; ev_arr[e] = ev;
    atomicAdd(&usum[u], eu);
    atomicAdd(&vsum[v], ev);
}

// ---------------------------------------------------------------------------
// t->q aggregation. u_idx is SORTED: accumulate runs of equal u in registers,
// flush atomics only on segment change (~32x fewer atomics). Note the reference
// quirk: it gathers Xt_val[u_idx] (not v_idx) — kept faithful.
__global__ __launch_bounds__(256) void scatter_u(
    const float* __restrict__ Xt_val, const float* __restrict__ eu_arr,
    const float* __restrict__ usum, const int* __restrict__ u_idx,
    float* __restrict__ Xt2q)
{
    const int  EPW  = 64; // edges per wave
    const int  wave = (blockIdx.x * blockDim.x + threadIdx.x) >> 5;
    const int  lane = threadIdx.x & 31;
    long e0 = (long)wave * EPW;
    if (e0 >= NE) return;
    long e1 = e0 + EPW; if (e1 > NE) e1 = NE;

    int ucur = u_idx[e0];
    float acc[8] = {0.f,0.f,0.f,0.f,0.f,0.f,0.f,0.f};
    for (long e = e0; e < e1; ++e) {
        const int u = u_idx[e];
        if (u != ucur) {
            float* op = Xt2q + (long)ucur * DD + lane * 8;
#pragma unroll
            for (int i = 0; i < 8; ++i) { atomicAdd(op + i, acc[i]); acc[i] = 0.f; }
            ucur = u;
        }
        const float a = eu_arr[e] / usum[u];
        const float4* vp = (const float4*)(Xt_val + (long)u * DD) + lane * 2;
        float4 v0 = vp[0], v1 = vp[1];
        acc[0] += a*v0.x; acc[1] += a*v0.y; acc[2] += a*v0.z; acc[3] += a*v0.w;
        acc[4] += a*v1.x; acc[5] += a*v1.y; acc[6] += a*v1.z; acc[7] += a*v1.w;
    }
    float* op = Xt2q + (long)ucur * DD + lane * 8;
#pragma unroll
    for (int i = 0; i < 8; ++i) atomicAdd(op + i, acc[i]);
}

// q->t aggregation: v_idx is random -> straight L2 atomics. Gathers Xq_val[u_idx].
__global__ __launch_bounds__(256) void scatter_v(
    const float* __restrict__ Xq_val, const float* __restrict__ ev_arr,
    const float* __restrict__ vsum,
    const int* __restrict__ u_idx, const int* __restrict__ v_idx,
    float* __restrict__ Xq2t)
{
    const int e    = (blockIdx.x * blockDim.x + threadIdx.x) >> 5;
    const int lane = threadIdx.x & 31;
    if (e >= NE) return;
    const int u = u_idx[e], v = v_idx[e];
    const float a = ev_arr[e] / vsum[v];
    const float4* vp = (const float4*)(Xq_val + (long)u * DD) + lane * 2;
    float4 v0 = vp[0], v1 = vp[1];
    float* op = Xq2t + (long)v * DD + lane * 8;
    atomicAdd(op + 0, a*v0.x); atomicAdd(op + 1, a*v0.y);
    atomicAdd(op + 2, a*v0.z); atomicAdd(op + 3, a*v0.w);
    atomicAdd(op + 4, a*v1.x); atomicAdd(op + 5, a*v1.y);
    atomicAdd(op + 6, a*v1.z); atomicAdd(op + 7, a*v1.w);
}

// column mean of Xq (for the broadcast Q row)
__global__ __launch_bounds__(256) void col_sum(const float* __restrict__ Xq,
                                               float* __restrict__ qacc)
{
    const int c  = threadIdx.x;        // 256 columns
    const int r0 = blockIdx.x * 200;   // 100 blocks x 200 rows
    float s = 0.f;
    for (int r = 0; r < 200; ++r) s += Xq[(long)(r0 + r) * DD + c];
    atomicAdd(&qacc[c], s);
}
__global__ void col_scale(const float* __restrict__ qacc, float* __restrict__ qrow)
{
    qrow[threadIdx.x] = qacc[threadIdx.x] * (1.0f / (float)NQ);
}

// ---------------------------------------------------------------------------
extern "C" void kernel_launch(void* const* d_in, const int* in_sizes, int n_in,
                              void* d_out, int out_size, void* d_ws, size_t ws_size,
                              hipStream_t stream)
{
    (void)in_sizes; (void)n_in; (void)out_size; (void)ws_size;

    const float* Xq      = (const float*)d_in[0];
    const float* Xt      = (const float*)d_in[1];
    const int*   u_idx   = (const int*)d_in[2];
    const int*   v_idx   = (const int*)d_in[3];
    const float* W_att_q = (const float*)d_in[4];
    const float* b_att_q = (const float*)d_in[5];
    const float* W_att_t = (const float*)d_in[6];
    const float* b_att_t = (const float*)d_in[7];
    const float* W_val_q = (const float*)d_in[8];
    const float* b_val_q = (const float*)d_in[9];
    const float* W_val_t = (const float*)d_in[10];
    const float* b_val_t = (const float*)d_in[11];
    const float* W_mrg_q = (const float*)d_in[12];
    const float* b_mrg_q = (const float*)d_in[13];
    const float* W_mrg_t = (const float*)d_in[14];
    const float* b_mrg_t = (const float*)d_in[15];

    float* out_q = (float*)d_out;                 // Xq_merged [NQ, D]
    float* out_t = out_q + (size_t)NQ * DD;       // Xt_merged [NT, D]

    // ---- workspace layout (floats) -------------------------------------
    float* w = (float*)d_ws;
    size_t off = 0;
    float* xq_att = w + off; off += (size_t)NQ * DD;
    float* xt_att = w + off; off += (size_t)NT * DD;
    float* xq_val = w + off; off += (size_t)NQ * DD;
    float* xt_val = w + off; off += (size_t)NT * DD;
    float* logits = w + off; off += (size_t)NE;
    float* eu_arr = w + off; off += (size_t)NE;
    float* ev_arr = w + off; off += (size_t)NE;
    float* qrow   = w + off; off += (size_t)DD;
    // ---- zero-initialized region (single capture-safe memset) ----------
    size_t zoff   = off;
    float* xt2q   = w + off; off += (size_t)NQ * DD;
    float* xq2t   = w + off; off += (size_t)NT * DD;
    float* usum   = w + off; off += (size_t)NQ;
    float* vsum   = w + off; off += (size_t)NT;
    float* qacc   = w + off; off += (size_t)DD;
    unsigned* umax = (unsigned*)(w + off); off += (size_t)NQ;   // 0 == -inf sentinel
    unsigned* vmax = (unsigned*)(w + off); off += (size_t)NT;
    hipMemsetAsync(w + zoff, 0, (off - zoff) * sizeof(float), stream);

    // ---- 1) four projection GEMMs with fused ELU (WMMA f32) ------------
    const int gemmBlocks = ((NQ / 16) * (DD / 16)) / 8;  // 8 waves / block
    gemm_wmma_f32<false,false,true ><<<gemmBlocks, 256, 0, stream>>>(
        Xq, nullptr, nullptr, W_att_q, b_att_q, xq_att);
    gemm_wmma_f32<false,false,true ><<<gemmBlocks, 256, 0, stream>>>(
        Xt, nullptr, nullptr, W_att_t, b_att_t, xt_att);
    gemm_wmma_f32<false,false,true ><<<gemmBlocks, 256, 0, stream>>>(
        Xq, nullptr, nullptr, W_val_q, b_val_q, xq_val);
    gemm_wmma_f32<false,false,true ><<<gemmBlocks, 256, 0, stream>>>(
        Xt, nullptr, nullptr, W_val_t, b_val_t, xt_val);

    // ---- 2) column mean of Xq (broadcast Q row) ------------------------
    col_sum<<<100, 256, 0, stream>>>(Xq, qacc);
    col_scale<<<1, 256, 0, stream>>>(qacc, qrow);

    // ---- 3) edge logits + fused segment max ----------------------------
    edge_logits<<<NE / 8, 256, 0, stream>>>(xq_att, xt_att, u_idx, v_idx,
                                            logits, umax, vmax);
    // ---- 4) exp + segment sums -----------------------------------------
    edge_expsum<<<NE / 256, 256, 0, stream>>>(logits, u_idx, v_idx, umax, vmax,
                                              eu_arr, ev_arr, usum, vsum);
    // ---- 5) aggregations -----------------------------------------------
    scatter_u<<<(NE / 64) / 8, 256, 0, stream>>>(xt_val, eu_arr, usum, u_idx, xt2q);
    scatter_v<<<NE / 8, 256, 0, stream>>>(xq_val, ev_arr, vsum, u_idx, v_idx, xq2t);

    // ---- 6) merge GEMMs (concat folded into multi-part A) --------------
    // Xq_merged = [Xq | Xt2q] @ W_mrg_q + b_mrg_q          (K = 512)
    gemm_wmma_f32<true ,false,false><<<gemmBlocks, 256, 0, stream>>>(
        Xq, xt2q, nullptr, W_mrg_q, b_mrg_q, out_q);
    // Xt_merged = [Xt | Xq2t | Qrow(bcast)] @ W_mrg_t + b  (K = 768)
    gemm_wmma_f32<true ,true ,false><<<gemmBlocks, 256, 0, stream>>>(
        Xt, xq2t, qrow, W_mrg_t, b_mrg_t, out_t);
}